// DetectionLoss_58471684768045
// MI455X (gfx1250) — compile-verified
//
#include <hip/hip_runtime.h>
#include <hip/hip_bf16.h>
#include <stdint.h>

#define BB 32
#define AA 65536
#define TT 32
#define CC 21
#define BLK 256

typedef __attribute__((ext_vector_type(2))) float v2f;
typedef __attribute__((ext_vector_type(8))) float v8f;
typedef __attribute__((ext_vector_type(4))) unsigned int u32x4;
typedef __attribute__((ext_vector_type(8))) int i32x8;
typedef __attribute__((ext_vector_type(4))) int i32x4;

// -------- IoU (matches reference: inter / (union + 1e-6)) --------
__device__ __forceinline__ float iou_box(float4 p, float4 q) {
    float x1 = fmaxf(p.x, q.x), y1 = fmaxf(p.y, q.y);
    float x2 = fminf(p.z, q.z), y2 = fminf(p.w, q.w);
    float inter = fmaxf(x2 - x1, 0.0f) * fmaxf(y2 - y1, 0.0f);
    float a1 = (p.z - p.x) * (p.w - p.y);
    float a2 = (q.z - q.x) * (q.w - q.y);
    return inter / (a1 + a2 - inter + 1e-6f);
}

// -------- wave32 sum reduction via V_WMMA_F32_16X16X4_F32 --------
// A = 16x4 f32 tile = 2 VGPRs/lane: lane m supplies A[m][0..1], lane m+16 supplies
// A[m][2..3]. With a = (v, 0) and B = ones, D[m][*] = v[m] + v[m+16]. Summing the
// 8 D VGPRs gives (lanes 0-15) S0..S7 and (lanes 16-31) S8..S15; one xor-16
// shuffle + add completes the 32-lane sum. 1 WMMA + 7 adds + 1 permute.
__device__ __forceinline__ float wave_sum_wmma(float v) {
    v2f a;  a.x = v;    a.y = 0.0f;
    v2f bo; bo.x = 1.0f; bo.y = 1.0f;
    v8f c = {};
    v8f d = __builtin_amdgcn_wmma_f32_16x16x4_f32(
        /*neg_a=*/false, a, /*neg_b=*/false, bo,
        /*c_mod=*/(short)0, c, /*reuse_a=*/false, /*reuse_b=*/false);
    float s = d[0] + d[1] + d[2] + d[3] + d[4] + d[5] + d[6] + d[7];
    s += __shfl_xor(s, 16, 32);
    return s;
}

// -------- kernel 0: zero per-sample accumulators + output --------
__global__ void k_init(unsigned long long* best0, unsigned* any_pos,
                       unsigned* num_pos, unsigned* neg_cnt,
                       float* pos_sum, float* bbox_sum, float* out) {
    int t = threadIdx.x;
    if (t < BB) {
        best0[t] = 0ull;  any_pos[t] = 0u;
        num_pos[t] = 0u;  neg_cnt[t] = 0u;
        pos_sum[t] = 0.0f; bbox_sum[t] = 0.0f;
    }
    if (t == 0) out[0] = 0.0f;
}

// -------- kernel 1: per-sample any-positive flag + argmax of ious[:,0] --------
__global__ void k_stats(const float* __restrict__ anchors,
                        const float* __restrict__ tboxes,
                        unsigned long long* __restrict__ best0,
                        unsigned* __restrict__ any_pos) {
    __shared__ float4 tb[TT];
    const int b = blockIdx.y;
    const int a = blockIdx.x * BLK + threadIdx.x;
    if (threadIdx.x < TT)
        tb[threadIdx.x] = ((const float4*)tboxes)[b * TT + threadIdx.x];
    __syncthreads();

    float4 an = ((const float4*)anchors)[a];
    float maxiou = 0.0f, iou0 = 0.0f;
    #pragma unroll 8
    for (int t = 0; t < TT; ++t) {
        float v = iou_box(an, tb[t]);
        if (t == 0) iou0 = v;
        maxiou = fmaxf(maxiou, v);
    }
    const int lane = threadIdx.x & 31;
    // any positive (>= 0.5)?
    unsigned long long bal = __ballot(maxiou >= 0.5f);
    if (lane == 0 && bal) atomicOr(&any_pos[b], 1u);
    // argmax of ious[:,0], first index wins ties: key = (bits(iou0), ~a)
    unsigned long long key =
        ((unsigned long long)__float_as_uint(iou0) << 32) |
        (unsigned long long)(0xFFFFFFFFu - (unsigned)a);
    #pragma unroll
    for (int off = 16; off > 0; off >>= 1) {
        unsigned long long o = __shfl_xor(key, off, 32);
        if (o > key) key = o;
    }
    if (lane == 0) atomicMax(&best0[b], key);
}

// -------- kernel 2: IoU recompute + CE (conf staged via TDM) + accumulate --------
__global__ void k_main(const float* __restrict__ bbox_pred,
                       const float* __restrict__ conf_pred,
                       const float* __restrict__ anchors,
                       const float* __restrict__ tboxes,
                       const int* __restrict__ tlabels,
                       const unsigned long long* __restrict__ best0,
                       const unsigned* __restrict__ any_pos,
                       float* __restrict__ neg_ce,
                       unsigned* __restrict__ num_pos,
                       unsigned* __restrict__ neg_cnt,
                       float* __restrict__ pos_sum,
                       float* __restrict__ bbox_sum) {
    __shared__ float lconf[BLK * CC];   // 21504 B staged tile of conf_pred
    __shared__ float4 tb[TT];
    __shared__ int tl[TT];

    const int b  = blockIdx.y;
    const int a0blk = blockIdx.x * BLK;
    const int a  = a0blk + threadIdx.x;

    if (threadIdx.x < TT) {
        tb[threadIdx.x] = ((const float4*)tboxes)[b * TT + threadIdx.x];
        tl[threadIdx.x] = tlabels[b * TT + threadIdx.x];
    }

    // ---- TDM: async DMA of the 256x21 f32 conf tile into LDS (wave 0 issues) ----
    if (threadIdx.x < 32) {
        unsigned long long gaddr =
            (unsigned long long)(uintptr_t)(conf_pred + ((size_t)b * AA + a0blk) * CC);
        unsigned lds_off = (unsigned)(uintptr_t)&lconf[0];  // flat addr[31:0] = LDS byte offset
        const unsigned NE = BLK * CC;                        // 5376 f32 elements
        u32x4 g0;
        g0[0] = 1u;                                          // count=1, user mode
        g0[1] = lds_off;                                     // lds_addr
        g0[2] = (unsigned)(gaddr & 0xFFFFFFFFull);           // global_addr[31:0]
        g0[3] = (unsigned)((gaddr >> 32) & 0x01FFFFFFull) | (2u << 30); // addr[56:32] | type=2
        i32x8 g1;
        g1[0] = (int)(2u << 16);                             // data_size = 4 bytes
        g1[1] = (int)((NE & 0xFFFFu) << 16);                 // tensor_dim0[15:0]
        g1[2] = (int)((NE >> 16) | (1u << 16));              // tensor_dim0 hi | tensor_dim1=1
        g1[3] = (int)((NE & 0xFFFFu) << 16);                 // tile_dim0 = 5376
        g1[4] = 1;                                           // tile_dim1 = 1
        g1[5] = (int)NE;                                     // tensor_dim0_stride
        g1[6] = 0; g1[7] = 0;
        i32x4 z4 = {0, 0, 0, 0};
        i32x8 z8 = {0, 0, 0, 0, 0, 0, 0, 0};
        __builtin_amdgcn_tensor_load_to_lds(g0, g1, z4, z4, z8, 0);
        __builtin_amdgcn_s_wait_tensorcnt(0);
    }
    __syncthreads();

    // ---- IoU row recompute: max + first-argmax ----
    float4 an = ((const float4*)anchors)[a];
    float maxiou = -1.0f; int matched = 0;
    #pragma unroll 8
    for (int t = 0; t < TT; ++t) {
        float v = iou_box(an, tb[t]);
        if (v > maxiou) { maxiou = v; matched = t; }
    }

    const unsigned long long bk = best0[b];
    const unsigned best_a = 0xFFFFFFFFu - (unsigned)(bk & 0xFFFFFFFFull);
    const bool forced = (any_pos[b] == 0u) && ((unsigned)a == best_a);
    const bool pos = (maxiou >= 0.5f) || forced;
    const bool neg = (maxiou < 0.4f) && !forced;
    if (forced) matched = 0;
    const int tgt = pos ? tl[matched] : 0;

    // ---- cross-entropy from staged LDS row ----
    const float* row = &lconf[threadIdx.x * CC];
    float m = row[0];
    #pragma unroll
    for (int c = 1; c < CC; ++c) m = fmaxf(m, row[c]);
    float s = 0.0f;
    #pragma unroll
    for (int c = 0; c < CC; ++c) s += __expf(row[c] - m);
    const float ce = m + __logf(s) - row[tgt];

    neg_ce[(size_t)b * AA + a] = neg ? ce : 0.0f;

    // ---- smooth L1 on positives ----
    float bbl = 0.0f;
    if (pos) {
        float4 bp = ((const float4*)bbox_pred)[(size_t)b * AA + a];
        float4 mb = tb[matched];
        float d0 = fabsf(bp.x - mb.x), d1 = fabsf(bp.y - mb.y);
        float d2 = fabsf(bp.z - mb.z), d3 = fabsf(bp.w - mb.w);
        bbl  = (d0 < 1.0f) ? 0.5f * d0 * d0 : d0 - 0.5f;
        bbl += (d1 < 1.0f) ? 0.5f * d1 * d1 : d1 - 0.5f;
        bbl += (d2 < 1.0f) ? 0.5f * d2 * d2 : d2 - 0.5f;
        bbl += (d3 < 1.0f) ? 0.5f * d3 * d3 : d3 - 0.5f;
    }

    // ---- per-wave accumulation: ballots for counts, WMMA for float sums ----
    const int lane = threadIdx.x & 31;
    unsigned long long bp_bal = __ballot(pos);
    unsigned long long bn_bal = __ballot(neg);
    float ps = wave_sum_wmma(pos ? ce : 0.0f);
    float bs = wave_sum_wmma(bbl);
    if (lane == 0) {
        if (bp_bal) atomicAdd(&num_pos[b], (unsigned)__popcll(bp_bal));
        if (bn_bal) atomicAdd(&neg_cnt[b], (unsigned)__popcll(bn_bal));
        atomicAdd(&pos_sum[b], ps);
        atomicAdd(&bbox_sum[b], bs);
    }
}

// -------- kernel 3: exact top-k sum via 4-level radix select on float bits --------
__global__ void k_select(const float* __restrict__ neg_ce,
                         const unsigned* __restrict__ num_pos,
                         const unsigned* __restrict__ neg_cnt,
                         const float* __restrict__ pos_sum,
                         const float* __restrict__ bbox_sum,
                         float* __restrict__ sample_loss) {
    __shared__ unsigned cnt[256];
    __shared__ float    sm[256];
    __shared__ unsigned sh_prefix, sh_k, sh_done;
    __shared__ float    sh_sum;

    const int b = blockIdx.x;
    const int tid = threadIdx.x;
    const unsigned np = num_pos[b];
    const unsigned nc = neg_cnt[b];
    unsigned k = 3u * np; if (nc < k) k = nc;

    if (tid == 0) { sh_prefix = 0u; sh_k = k; sh_sum = 0.0f; sh_done = (k == 0u); }
    __syncthreads();

    for (int level = 3; level >= 0; --level) {
        if (sh_done) break;                 // uniform across block
        const int shift = level * 8;
        cnt[tid] = 0u; sm[tid] = 0.0f;
        __syncthreads();
        const unsigned prefix = sh_prefix;
        for (int i = tid; i < AA; i += BLK) {
            unsigned bits = __float_as_uint(neg_ce[(size_t)b * AA + i]);
            if (level == 3 || (bits >> (shift + 8)) == prefix) {
                unsigned bin = (bits >> shift) & 0xFFu;
                atomicAdd(&cnt[bin], 1u);
                atomicAdd(&sm[bin], __uint_as_float(bits));
            }
        }
        __syncthreads();
        if (tid == 0) {
            unsigned kr = sh_k; float acc = sh_sum;
            for (int bin = 255; bin >= 0; --bin) {
                unsigned c = cnt[bin];
                if (c < kr)      { kr -= c; acc += sm[bin]; }
                else if (c == kr) { acc += sm[bin]; kr = 0u; sh_done = 1u; break; }
                else {
                    if (level == 0) {       // exact value; take kr copies
                        float v = __uint_as_float((prefix << 8) | (unsigned)bin);
                        acc += (float)kr * v; kr = 0u; sh_done = 1u;
                    } else {
                        sh_prefix = (prefix << 8) | (unsigned)bin;
                    }
                    break;
                }
            }
            sh_k = kr; sh_sum = acc;
        }
        __syncthreads();
    }

    if (tid == 0) {
        float neg_sum = sh_sum;
        float cl = (pos_sum[b] + neg_sum) / (float)(np + k);
        float bl = bbox_sum[b] / (float)np;   // np >= 1 by forcing
        sample_loss[b] = cl + bl;
    }
}

// -------- kernel 4: deterministic final mean --------
__global__ void k_final(const float* __restrict__ sample_loss, float* __restrict__ out) {
    if (threadIdx.x == 0) {
        float s = 0.0f;
        for (int b = 0; b < BB; ++b) s += sample_loss[b];
        out[0] = s / (float)BB;
    }
}

extern "C" void kernel_launch(void* const* d_in, const int* in_sizes, int n_in,
                              void* d_out, int out_size, void* d_ws, size_t ws_size,
                              hipStream_t stream) {
    const float* bbox_pred = (const float*)d_in[0];   // [B,A,4]
    const float* conf_pred = (const float*)d_in[1];   // [B,A,C]
    const float* anchors   = (const float*)d_in[2];   // [A,4]
    const float* tboxes    = (const float*)d_in[3];   // [B,T,4]
    const int*   tlabels   = (const int*)d_in[4];     // [B,T]
    float* out = (float*)d_out;

    // workspace layout
    char* base = (char*)d_ws;
    const size_t NEG_BYTES = (size_t)BB * AA * sizeof(float);    // 8 MiB
    float* neg_ce = (float*)base;
    unsigned long long* best0 = (unsigned long long*)(base + NEG_BYTES);
    unsigned* any_pos = (unsigned*)(best0 + BB);
    unsigned* num_pos = any_pos + BB;
    unsigned* neg_cnt = num_pos + BB;
    float* pos_sum  = (float*)(neg_cnt + BB);
    float* bbox_sum = pos_sum + BB;
    float* sample_loss = bbox_sum + BB;

    dim3 grid(AA / BLK, BB);

    k_init<<<1, BLK, 0, stream>>>(best0, any_pos, num_pos, neg_cnt,
                                  pos_sum, bbox_sum, out);
    k_stats<<<grid, BLK, 0, stream>>>(anchors, tboxes, best0, any_pos);
    k_main<<<grid, BLK, 0, stream>>>(bbox_pred, conf_pred, anchors, tboxes, tlabels,
                                     best0, any_pos, neg_ce,
                                     num_pos, neg_cnt, pos_sum, bbox_sum);
    k_select<<<BB, BLK, 0, stream>>>(neg_ce, num_pos, neg_cnt,
                                     pos_sum, bbox_sum, sample_loss);
    k_final<<<1, 32, 0, stream>>>(sample_loss, out);
}